// PenalizedMSELoss_73538430042130
// MI455X (gfx1250) — compile-verified
//
#include <hip/hip_runtime.h>
#include <hip/hip_bf16.h>

// PenalizedMSELoss on MI455X (gfx1250): pure bandwidth-bound streaming
// reduction. 268 MB read @ 23.3 TB/s => ~11.5us floor. Strategy:
//   * b128 non-temporal vector loads (working set 256MB > 192MB L2)
//   * fp32 tree accumulation, deterministic two-pass reduction (no atomics)
//   * wave32 cross-lane sum via V_WMMA_F32_16X16X4_F32 (A = lane partials,
//     B = ones) + one xor-16 shuffle.

typedef __attribute__((ext_vector_type(2))) float v2f;
typedef __attribute__((ext_vector_type(4))) float v4f;
typedef __attribute__((ext_vector_type(4))) int   v4i;
typedef __attribute__((ext_vector_type(8))) float v8f;

#define K_LABEL 5
#define K_LOWER 4.5f
#define K_UPPER 5.5f
#define K_PW    3.0f

__device__ __forceinline__ float penal_elem(float x, int t) {
    float diff = x - (float)t;
    bool penal = (x > K_LOWER) & (x < K_UPPER) & (t != K_LABEL);
    float w = penal ? K_PW : 1.0f;
    return w * diff * diff;
}

// Sum the 32 per-lane fp32 partials of a wave using one WMMA.
// A (16x4 f32, 2 VGPRs/lane) holds each lane's partial exactly once and zeros
// elsewhere; B (4x16) is all ones regardless of register layout. Then
// D[m][n] = rowsum_A(m), so any full column of D sums to the wave total.
// Column n lives in lanes n (M=0..7 across the 8 D VGPRs) and n+16 (M=8..15),
// so per-lane sum of the 8 D regs + xor-16 shuffle = total, in every lane.
__device__ __forceinline__ float wave_sum_wmma(float p) {
    v2f a; a[0] = p;    a[1] = 0.0f;
    v2f b; b[0] = 1.0f; b[1] = 1.0f;
    v8f c = {};
    v8f d = __builtin_amdgcn_wmma_f32_16x16x4_f32(
        /*neg_a=*/false, a, /*neg_b=*/false, b,
        /*c_mod=*/(short)0, c, /*reuse_a=*/false, /*reuse_b=*/false);
    float s = ((d[0] + d[1]) + (d[2] + d[3])) + ((d[4] + d[5]) + (d[6] + d[7]));
    s += __shfl_xor(s, 16, 32);
    return s;
}

// Block-level deterministic tree sum. Valid result in threadIdx.x == 0.
__device__ __forceinline__ float block_sum(float v) {
    __shared__ float ws[32];               // up to 32 waves per block
    float s = wave_sum_wmma(v);            // EXEC all-1s here: no divergence
    int wid  = threadIdx.x >> 5;
    int lane = threadIdx.x & 31;
    if (lane == 0) ws[wid] = s;
    __syncthreads();
    float tot = 0.0f;
    if (threadIdx.x == 0) {
        int nw = blockDim.x >> 5;
        for (int i = 0; i < nw; ++i) tot += ws[i];
    }
    return tot;
}

__global__ void penal_mse_partial_kernel(const float* __restrict__ x,
                                         const int* __restrict__ t,
                                         float* __restrict__ partial,
                                         long long n) {
    const long long n4 = n >> 2;
    const v4f* __restrict__ x4 = (const v4f*)x;
    const v4i* __restrict__ t4 = (const v4i*)t;

    float acc = 0.0f;
    const long long stride = (long long)gridDim.x * blockDim.x;
    for (long long i = (long long)blockIdx.x * blockDim.x + threadIdx.x;
         i < n4; i += stride) {
        v4f xv = __builtin_nontemporal_load(&x4[i]);   // global_load_b128, TH=NT
        v4i tv = __builtin_nontemporal_load(&t4[i]);
        acc += penal_elem(xv[0], tv[0]);
        acc += penal_elem(xv[1], tv[1]);
        acc += penal_elem(xv[2], tv[2]);
        acc += penal_elem(xv[3], tv[3]);
    }
    // Scalar tail (N = 2^25 has none, but stay general). Re-converges before
    // the WMMA reduction so EXEC is all-1s there.
    if (blockIdx.x == 0 && threadIdx.x == 0) {
        for (long long i = n4 << 2; i < n; ++i) acc += penal_elem(x[i], t[i]);
    }

    float b = block_sum(acc);
    if (threadIdx.x == 0) partial[blockIdx.x] = b;
}

__global__ void penal_mse_final_kernel(const float* __restrict__ partial,
                                       float* __restrict__ out,
                                       int nblocks, float inv_n) {
    float acc = 0.0f;
    for (int i = threadIdx.x; i < nblocks; i += blockDim.x) acc += partial[i];
    float b = block_sum(acc);
    if (threadIdx.x == 0) out[0] = b * inv_n;
}

extern "C" void kernel_launch(void* const* d_in, const int* in_sizes, int n_in,
                              void* d_out, int out_size, void* d_ws, size_t ws_size,
                              hipStream_t stream) {
    const float* x = (const float*)d_in[0];
    const int*   t = (const int*)d_in[1];   // jnp "int64" materializes as int32
    float* out     = (float*)d_out;
    float* partial = (float*)d_ws;

    const long long n = (long long)in_sizes[0];
    const int threads = 256;                 // 8 wave32s / block

    // Enough blocks to saturate HBM, bounded by work and scratch capacity.
    long long n4     = n >> 2;
    long long needed = (n4 + threads - 1) / threads;
    long long blocks = 4096;
    if (blocks > needed) blocks = needed > 0 ? needed : 1;
    long long ws_cap = (long long)(ws_size / sizeof(float));
    if (ws_cap > 0 && blocks > ws_cap) blocks = ws_cap;

    penal_mse_partial_kernel<<<(int)blocks, threads, 0, stream>>>(x, t, partial, n);
    penal_mse_final_kernel<<<1, threads, 0, stream>>>(partial, out, (int)blocks,
                                                      1.0f / (float)n);
}